// catcher_51814485458995
// MI455X (gfx1250) — compile-verified
//
#include <hip/hip_runtime.h>

// ---------------------------------------------------------------------------
// Problem constants (from the reference)
// ---------------------------------------------------------------------------
#define BATCH   64
#define NPATCH  162
#define CDIM    2048
#define LDS_S   164                         // padded row stride for 162-wide score mats (16B aligned rows)
#define ATTN_SCALE 0.022097086912079612f    // 2048^-0.5
#define NORM_EPS   1e-12f

typedef __attribute__((ext_vector_type(2))) float v2f;
typedef __attribute__((ext_vector_type(4))) float v4f;
typedef __attribute__((ext_vector_type(8))) float v8f;
typedef __attribute__((ext_vector_type(4))) unsigned int v4u;
typedef __attribute__((ext_vector_type(4))) int v4i;
typedef __attribute__((ext_vector_type(8))) int v8i;

// ---------------------------------------------------------------------------
// Tensor Data Mover support (gfx1250). amdgpu-toolchain (clang-23) 6-arg form:
//   (uint32x4 g0, int32x8 g1, int32x4 g2, int32x4 g3, int32x8, i32 cpol)
// ---------------------------------------------------------------------------
#if defined(__has_builtin)
#if __has_builtin(__builtin_amdgcn_tensor_load_to_lds) && __has_builtin(__builtin_amdgcn_s_wait_tensorcnt)
#define USE_TDM 1
#endif
#endif
#ifndef USE_TDM
#define USE_TDM 0
#endif

#if USE_TDM
// 2-D f32 tile load global->LDS via TDM.
//   tileX elements contiguous per row, tileY rows, row stride = strideElems.
//   LDS rows padded: +1 DWORD after every 32 DWORDs (pad_interval=4, pad_amount=0)
//   -> matches a [Y][33]-float LDS array when tileX == 32.
__device__ __forceinline__ void tdm_load_tile_f32(unsigned ldsOff, const float* gptr,
                                                  unsigned tileX, unsigned tileY,
                                                  unsigned strideElems) {
    const unsigned long long ga = (unsigned long long)(size_t)gptr;
    v4u g0;
    g0[0] = 1u;                                              // count=1, user D#
    g0[1] = ldsOff;                                          // lds_addr (bytes)
    g0[2] = (unsigned)(ga & 0xFFFFFFFFu);                    // global_addr[31:0]
    g0[3] = (unsigned)((ga >> 32) & 0x01FFFFFFu) | 0x80000000u; // global_addr[56:32] | type=2
    v8i g1;
    g1[0] = (int)((2u << 16) | (1u << 20) | (4u << 22));     // data_size=4B, pad_en, pad every 32 DW (+1 DW)
    g1[1] = (int)((tileX & 0xFFFFu) << 16);                  // tensor_dim0[15:0] @ bits 63:48
    g1[2] = (int)(((tileX >> 16) & 0xFFFFu) | ((tileY & 0xFFFFu) << 16)); // dim0 hi | tensor_dim1 lo
    g1[3] = (int)(((tileY >> 16) & 0xFFFFu) | ((tileX & 0xFFFFu) << 16)); // dim1 hi | tile_dim0
    g1[4] = (int)(tileY & 0xFFFFu);                          // tile_dim1 | tile_dim2=0
    g1[5] = (int)strideElems;                                // tensor_dim0_stride[31:0]
    g1[6] = 0;                                               // stride hi | dim1_stride lo (unused, 2D)
    g1[7] = 0;
    v4i z4 = {};
    v8i z8 = {};
    __builtin_amdgcn_tensor_load_to_lds(g0, g1, z4, z4, z8, 0);
}
#endif

// ---------------------------------------------------------------------------
// Row inverse-norm: rnorm[row] = 1 / max(||g[row,:]||, eps). One wave per row.
// ---------------------------------------------------------------------------
__global__ __launch_bounds__(32)
void rownorm_kernel(const float* __restrict__ g, float* __restrict__ rnorm) {
    const size_t row = blockIdx.x;
    const float* p = g + row * (size_t)CDIM;
    const int lane = threadIdx.x;
    float s = 0.0f;
    #pragma unroll
    for (int t = 0; t < 16; ++t) {
        v4f v = *(const v4f*)(p + (((t << 5) + lane) << 2));
        s += v.x * v.x + v.y * v.y + v.z * v.z + v.w * v.w;
    }
    #pragma unroll
    for (int o = 16; o > 0; o >>= 1) s += __shfl_xor(s, o, 32);
    if (lane == 0) rnorm[row] = 1.0f / fmaxf(sqrtf(s), NORM_EPS);
}

// ---------------------------------------------------------------------------
// In-place row softmax over n valid columns (row stride ld). One wave per row.
// ---------------------------------------------------------------------------
__global__ __launch_bounds__(32)
void softmax_rows_kernel(float* __restrict__ X, int n, int ld) {
    float* p = X + (size_t)blockIdx.x * (size_t)ld;
    const int lane = threadIdx.x;
    float v[6];
    float mx = -3.0e38f;
    #pragma unroll
    for (int i = 0; i < 6; ++i) {
        int j = lane + (i << 5);
        v[i] = (j < n) ? p[j] : -3.0e38f;
        mx = fmaxf(mx, v[i]);
    }
    #pragma unroll
    for (int o = 16; o > 0; o >>= 1) mx = fmaxf(mx, __shfl_xor(mx, o, 32));
    float s = 0.0f;
    #pragma unroll
    for (int i = 0; i < 6; ++i) {
        int j = lane + (i << 5);
        float e = (j < n) ? __expf(v[i] - mx) : 0.0f;
        v[i] = e; s += e;
    }
    #pragma unroll
    for (int o = 16; o > 0; o >>= 1) s += __shfl_xor(s, o, 32);
    const float inv = 1.0f / s;
    #pragma unroll
    for (int i = 0; i < 6; ++i) {
        int j = lane + (i << 5);
        if (j < n) p[j] = v[i] * inv;
    }
}

// ---------------------------------------------------------------------------
// Batched fp32 WMMA GEMM:  D[b] = scale( A[b] * B[b] )
//   TRANSB=0: Bmat[k][n] = B[k*ldb+n]    TRANSB=1: Bmat[k][n] = B[n*ldb+k]
//   SCALE_MODE 0: none  1: D *= rnorm[row]*rnorm[col]  2: D *= alpha
//   ADD_POS: A[i][k] += pos[i*lda+k]
// 4 wave32 per block; each wave owns a 16x16 C tile (block tile 64x16).
// K panels of 32 staged via TDM (interior tiles) or branch-free clamped loads.
// ---------------------------------------------------------------------------
template <int TRANSB, int SCALE_MODE, bool ADD_POS>
__global__ __launch_bounds__(128)
void gemm_wmma_f32(const float* __restrict__ A, long long strideA, int lda,
                   const float* __restrict__ Bm, long long strideB, int ldb,
                   float* __restrict__ D, long long strideD, int ldd,
                   int M, int Ndim, int K,
                   const float* __restrict__ rn,
                   const float* __restrict__ posm,
                   float alpha) {
    __shared__ float As[64][33];    // [m][k] panel, stride 33 (TDM pad-compatible)
    __shared__ float BsT[16][33];   // [n][k]

    const int b = blockIdx.z;
    A += (size_t)b * (size_t)strideA;
    Bm += (size_t)b * (size_t)strideB;
    D += (size_t)b * (size_t)strideD;
    if (SCALE_MODE == 1) rn += (size_t)b * NPATCH;

    const int m0 = blockIdx.x * 64;
    const int n0 = blockIdx.y * 16;
    const int t = threadIdx.x;
    const int lane = t & 31;
    const int wave = t >> 5;

    const bool tileAInterior = (!ADD_POS) && (m0 + 64 <= M);
    const bool tileBInterior = (TRANSB != 0) && (n0 + 16 <= Ndim);
#if USE_TDM
    const unsigned ldsOffA = (unsigned)(size_t)&As[0][0];
    const unsigned ldsOffB = (unsigned)(size_t)&BsT[0][0];
#endif

    v8f c = {};

    for (int k0 = 0; k0 < K; k0 += 32) {
        const bool kFull = (k0 + 32 <= K);
#if USE_TDM
        const bool tdmA = tileAInterior && kFull;
        const bool tdmB = tileBInterior && kFull;
        if (wave == 0) {
            if (tdmA) tdm_load_tile_f32(ldsOffA, A + (size_t)m0 * lda + k0, 32u, 64u, (unsigned)lda);
            if (tdmB) tdm_load_tile_f32(ldsOffB, Bm + (size_t)n0 * ldb + k0, 32u, 16u, (unsigned)ldb);
        }
#else
        const bool tdmA = false, tdmB = false;
        (void)tileAInterior; (void)tileBInterior; (void)kFull;
#endif
        // ---- A panel (branch-free clamped loads when TDM not applicable) ----
        if (!tdmA) {
            const int mrow = t >> 3;        // 0..15
            const int kq = (t & 7) << 2;    // 0,4,..,28
            const int gk = k0 + kq;
            #pragma unroll
            for (int mi = 0; mi < 4; ++mi) {
                const int m = (mi << 4) + mrow;
                const int gr = m0 + m;
                const bool rok = gr < M;
                const size_t rbase = (size_t)(rok ? gr : (M - 1)) * lda;
                if (gk + 4 <= K) {
                    v4f v = *(const v4f*)(A + rbase + gk);
                    if (ADD_POS) v = v + *(const v4f*)(posm + rbase + gk);
                    #pragma unroll
                    for (int e = 0; e < 4; ++e) As[m][kq + e] = rok ? v[e] : 0.0f;
                } else {
                    #pragma unroll
                    for (int e = 0; e < 4; ++e) {
                        const int ke = gk + e;
                        const int kec = ke < K ? ke : (K - 1);
                        float x = A[rbase + kec];
                        if (ADD_POS) x += posm[rbase + kec];
                        As[m][kq + e] = (rok && ke < K) ? x : 0.0f;
                    }
                }
            }
        }
        // ---- B panel ----
        if (!tdmB) {
            if (TRANSB) {
                const int n = t >> 3;       // 0..15
                const int kq = (t & 7) << 2;
                const int gn = n0 + n;
                const bool nok = gn < Ndim;
                const size_t nbase = (size_t)(nok ? gn : (Ndim - 1)) * ldb;
                const int gk = k0 + kq;
                if (gk + 4 <= K) {
                    v4f v = *(const v4f*)(Bm + nbase + gk);
                    #pragma unroll
                    for (int e = 0; e < 4; ++e) BsT[n][kq + e] = nok ? v[e] : 0.0f;
                } else {
                    #pragma unroll
                    for (int e = 0; e < 4; ++e) {
                        const int ke = gk + e;
                        const int kec = ke < K ? ke : (K - 1);
                        float x = Bm[nbase + kec];
                        BsT[n][kq + e] = (nok && ke < K) ? x : 0.0f;
                    }
                }
            } else {
                const int n = t & 15;
                const int kr = t >> 4;      // 0..7
                const int gn = n0 + n;
                const bool nok = gn < Ndim;
                const int gnc = nok ? gn : (Ndim - 1);
                #pragma unroll
                for (int kqi = 0; kqi < 4; ++kqi) {
                    const int k = (kqi << 3) + kr;
                    const int gk = k0 + k;
                    const int gkc = gk < K ? gk : (K - 1);
                    const float x = Bm[(size_t)gkc * ldb + gnc];
                    BsT[n][k] = (nok && gk < K) ? x : 0.0f;
                }
            }
        }
#if USE_TDM
        if (tdmA || tdmB) __builtin_amdgcn_s_wait_tensorcnt((short)0);
#endif
        __syncthreads();

        // ---- 8 x V_WMMA_F32_16X16X4_F32 ----
        const int mrowF = (wave << 4) + (lane & 15);
        const int kHi = (lane >> 4) << 1;   // 0 (lanes 0-15) / 2 (lanes 16-31)
        const int nF = lane & 15;
        #pragma unroll
        for (int kk = 0; kk < 8; ++kk) {
            const int kb = (kk << 2) + kHi;
            v2f a;  a.x = As[mrowF][kb];  a.y = As[mrowF][kb + 1];
            v2f bf; bf.x = BsT[nF][kb];  bf.y = BsT[nF][kb + 1];
            c = __builtin_amdgcn_wmma_f32_16x16x4_f32(
                    false, a, false, bf, (short)0, c, false, false);
        }
        __syncthreads();
    }

    // ---- epilogue: VGPR r -> M = r (+8 for lanes 16-31), N = lane&15 ----
    const int col = n0 + (lane & 15);
    const int rowBase = m0 + (wave << 4) + ((lane >> 4) << 3);
    #pragma unroll
    for (int r = 0; r < 8; ++r) {
        const int row = rowBase + r;
        if (row < M && col < Ndim) {
            float v = c[r];
            if (SCALE_MODE == 1) v *= rn[row] * rn[col];
            else if (SCALE_MODE == 2) v *= alpha;
            D[(size_t)row * ldd + col] = v;
        }
    }
}

// ---------------------------------------------------------------------------
// Launcher
// ---------------------------------------------------------------------------
extern "C" void kernel_launch(void* const* d_in, const int* in_sizes, int n_in,
                              void* d_out, int out_size, void* d_ws, size_t ws_size,
                              hipStream_t stream) {
    (void)in_sizes; (void)n_in; (void)out_size; (void)ws_size;
    const float* q   = (const float*)d_in[0];   // [64,162,2048]
    const float* g   = (const float*)d_in[1];   // [64,162,2048]
    const float* Wq  = (const float*)d_in[2];   // [2048,2048]
    const float* Wg  = (const float*)d_in[3];   // [2048,2048]
    const float* pos = (const float*)d_in[4];   // [1,162,2048]
    float* out = (float*)d_out;                 // [64,162,2048]
    float* ws = (float*)d_ws;

    const long long BNC = (long long)BATCH * NPATCH * CDIM;
    const long long sNC = (long long)NPATCH * CDIM;          // per-batch stride, big mats
    const long long sSS = (long long)NPATCH * LDS_S;         // per-batch stride, score mats

    float* rnorm = ws;                                   // 64*162
    float* S     = rnorm + (size_t)BATCH * NPATCH;       // 64*162*164 (scores, reused twice)
    float* relq  = S + (size_t)BATCH * NPATCH * LDS_S;   // 64*162*2048
    float* g2n   = relq + (size_t)BNC;                   // 64*162*2048
    float* relg  = g2n + (size_t)BNC;                    // 64*162*2048

    const dim3 blk(128);
    const dim3 gridBig(3, CDIM / 16, BATCH);             // 64x16 tiles over 162 x 2048
    const dim3 gridSmall(3, (NPATCH + 15) / 16, BATCH);  // over 162 x 162

    // 1) per-row inverse norms of g
    rownorm_kernel<<<BATCH * NPATCH, 32, 0, stream>>>(g, rnorm);

    // 2) cosine-similarity gram: S = (g gT) * rn_i rn_j
    gemm_wmma_f32<1, 1, false><<<gridSmall, blk, 0, stream>>>(
        g, sNC, CDIM, g, sNC, CDIM, S, sSS, LDS_S,
        NPATCH, NPATCH, CDIM, rnorm, nullptr, 1.0f);

    // 3) rel = softmax(S) rows, in place
    softmax_rows_kernel<<<BATCH * NPATCH, 32, 0, stream>>>(S, NPATCH, LDS_S);

    // 4) g2 = rel @ g
    gemm_wmma_f32<0, 0, false><<<gridBig, blk, 0, stream>>>(
        S, sSS, LDS_S, g, sNC, CDIM, g2n, sNC, CDIM,
        NPATCH, CDIM, NPATCH, nullptr, nullptr, 1.0f);

    // 5) rel_q = q @ WqT   (weight batch-stride 0)
    gemm_wmma_f32<1, 0, false><<<gridBig, blk, 0, stream>>>(
        q, sNC, CDIM, Wq, 0, CDIM, relq, sNC, CDIM,
        NPATCH, CDIM, CDIM, nullptr, nullptr, 1.0f);

    // 6) rel_g = (g2 + pos) @ WgT   (pos fused into A stage)
    gemm_wmma_f32<1, 0, true><<<gridBig, blk, 0, stream>>>(
        g2n, sNC, CDIM, Wg, 0, CDIM, relg, sNC, CDIM,
        NPATCH, CDIM, CDIM, nullptr, pos, 1.0f);

    // 7) logits = rel_q @ rel_gT * SCALE  -> reuse S buffer
    gemm_wmma_f32<1, 2, false><<<gridSmall, blk, 0, stream>>>(
        relq, sNC, CDIM, relg, sNC, CDIM, S, sSS, LDS_S,
        NPATCH, NPATCH, CDIM, nullptr, nullptr, ATTN_SCALE);

    // 8) attn = softmax(logits) rows, in place
    softmax_rows_kernel<<<BATCH * NPATCH, 32, 0, stream>>>(S, NPATCH, LDS_S);

    // 9) out = attn @ g2
    gemm_wmma_f32<0, 0, false><<<gridBig, blk, 0, stream>>>(
        S, sSS, LDS_S, g2n, sNC, CDIM, out, sNC, CDIM,
        NPATCH, CDIM, NPATCH, nullptr, nullptr, 1.0f);
}